// StructureExtractor_7842610283390
// MI455X (gfx1250) — compile-verified
//
#include <hip/hip_runtime.h>

#define NN 100000
#define NE 1250000
#define D 64
#define ED 16
#define NL 3
#define INNER 256            // (NL+1)*D
#define BN_EPS 1e-5f

typedef __attribute__((ext_vector_type(2))) float v2f;
typedef __attribute__((ext_vector_type(8))) float v8f;

__device__ __forceinline__ v8f wmma4(v2f a, v2f b, v8f c) {
  // D = A(16x4,f32) * B(4x16,f32) + C(16x16,f32)
  return __builtin_amdgcn_wmma_f32_16x16x4_f32(false, a, false, b, (short)0, c, false, false);
}

__device__ __forceinline__ v8f splat8(float v) {
  v8f r = {v, v, v, v, v, v, v, v};
  return r;
}

// ---------------- setup kernels ----------------

__global__ void k_init(float* __restrict__ deg, float* __restrict__ bnsum,
                       float* __restrict__ bnsumsq) {
  int i = blockIdx.x * blockDim.x + threadIdx.x;
  if (i < NN) deg[i] = 1.0f;                 // deg = count + 1
  if (i < INNER) { bnsum[i] = 0.0f; bnsumsq[i] = 0.0f; }
}

__global__ void k_degree(const int* __restrict__ row, float* __restrict__ deg) {
  int e = blockIdx.x * blockDim.x + threadIdx.x;
  if (e < NE) atomicAdd(&deg[row[e]], 1.0f);
}

__global__ void k_norm(const int* __restrict__ row, const int* __restrict__ col,
                       const float* __restrict__ deg, float* __restrict__ norm) {
  int e = blockIdx.x * blockDim.x + threadIdx.x;
  if (e < NE) norm[e] = rsqrtf(deg[row[e]]) * rsqrtf(deg[col[e]]);
}

__global__ void k_copy_x(const float* __restrict__ x, float* __restrict__ xcat) {
  int i = blockIdx.x * blockDim.x + threadIdx.x;
  if (i < NN * D) {
    int n = i >> 6, c = i & 63;
    xcat[(size_t)n * INNER + c] = x[i];
  }
}

// ---------------- per-layer: node GEMM + self term ----------------
// h = x_l @ lin_w[l] + lin_b[l]; out_init = relu(h + root_emb[l]) / deg

__global__ __launch_bounds__(128) void k_node_gemm(
    const float* __restrict__ xcat, const float* __restrict__ lin_w,
    const float* __restrict__ lin_b, const float* __restrict__ root_emb,
    const float* __restrict__ deg, float* __restrict__ h,
    float* __restrict__ outbuf, int layer) {
  int wave = (int)((blockIdx.x * blockDim.x + threadIdx.x) >> 5);
  int lane = threadIdx.x & 31;
  if (wave >= NN / 16) return;               // wave-uniform: EXEC stays full
  int mbase = wave * 16;
  int n = lane & 15;                          // A-row m / B,C column n
  int half = lane >> 4;

  const float* W  = lin_w + (size_t)layer * D * D;
  const float* Bv = lin_b + layer * D;
  const float* RE = root_emb + layer * D;

  v8f acc[4];
#pragma unroll
  for (int nt = 0; nt < 4; ++nt) acc[nt] = splat8(Bv[nt * 16 + n]);

  const float* Abase = xcat + (size_t)(mbase + n) * INNER + layer * D + 2 * half;
#pragma unroll 4
  for (int k = 0; k < 16; ++k) {             // K = 64 in steps of 4
    v2f a; a.x = Abase[4 * k]; a.y = Abase[4 * k + 1];
    int krow = 4 * k + 2 * half;
#pragma unroll
    for (int nt = 0; nt < 4; ++nt) {
      v2f b;
      b.x = W[(size_t)krow * D + nt * 16 + n];
      b.y = W[(size_t)(krow + 1) * D + nt * 16 + n];
      acc[nt] = wmma4(a, b, acc[nt]);
    }
  }

  float invd[8];
#pragma unroll
  for (int r = 0; r < 8; ++r) invd[r] = 1.0f / deg[mbase + r + 8 * half];

#pragma unroll
  for (int nt = 0; nt < 4; ++nt) {
    int ch = nt * 16 + n;
    float re = RE[ch];
#pragma unroll
    for (int r = 0; r < 8; ++r) {
      int m = r + 8 * half;
      float v = acc[nt][r];
      h[(size_t)(mbase + m) * D + ch] = v;
      float o = v + re;
      o = o > 0.0f ? o : 0.0f;
      outbuf[(size_t)(mbase + m) * D + ch] = o * invd[r];
    }
  }
}

// ---------------- per-layer: edge GEMM + gather + scatter-add ----------------
// ee = edge_attr @ edge_w[l] + edge_b[l]; msg = norm*relu(h[row]+ee); out[col]+=msg

__global__ __launch_bounds__(128) void k_edge(
    const float* __restrict__ edge_attr, const float* __restrict__ edge_w,
    const float* __restrict__ edge_b, const int* __restrict__ row,
    const int* __restrict__ col, const float* __restrict__ norm,
    const float* __restrict__ h, float* __restrict__ outbuf, int layer) {
  int wave = (int)((blockIdx.x * blockDim.x + threadIdx.x) >> 5);
  int lane = threadIdx.x & 31;
  if (wave >= NE / 16) return;
  int ebase = wave * 16;
  int n = lane & 15;
  int half = lane >> 4;

  const float* W  = edge_w + (size_t)layer * ED * D;
  const float* Bv = edge_b + layer * D;

  v8f acc[4];
#pragma unroll
  for (int nt = 0; nt < 4; ++nt) acc[nt] = splat8(Bv[nt * 16 + n]);

  const float* Abase = edge_attr + (size_t)(ebase + n) * ED + 2 * half;
#pragma unroll
  for (int k = 0; k < 4; ++k) {              // K = 16 in steps of 4
    v2f a; a.x = Abase[4 * k]; a.y = Abase[4 * k + 1];
    int krow = 4 * k + 2 * half;
#pragma unroll
    for (int nt = 0; nt < 4; ++nt) {
      v2f b;
      b.x = W[(size_t)krow * D + nt * 16 + n];
      b.y = W[(size_t)(krow + 1) * D + nt * 16 + n];
      acc[nt] = wmma4(a, b, acc[nt]);
    }
  }

  int rowv[8], colv[8];
  float nrm[8];
#pragma unroll
  for (int r = 0; r < 8; ++r) {
    int e = ebase + r + 8 * half;
    rowv[r] = row[e]; colv[r] = col[e]; nrm[r] = norm[e];
  }

#pragma unroll
  for (int nt = 0; nt < 4; ++nt) {
    int ch = nt * 16 + n;
#pragma unroll
    for (int r = 0; r < 8; ++r) {
      float v = acc[nt][r] + h[(size_t)rowv[r] * D + ch];
      v = v > 0.0f ? v * nrm[r] : 0.0f;
      if (v > 0.0f)                            // relu kills ~half the atomics
        atomicAdd(&outbuf[(size_t)colv[r] * D + ch], v);
    }
  }
}

// ---------------- relu + write into concat slot ----------------

__global__ void k_relu_copy(const float* __restrict__ outbuf,
                            float* __restrict__ xcat, int layer) {
  int i = blockIdx.x * blockDim.x + threadIdx.x;
  if (i < NN * D) {
    int nrow = i >> 6, c = i & 63;
    float v = outbuf[i];
    xcat[(size_t)nrow * INNER + (layer + 1) * D + c] = v > 0.0f ? v : 0.0f;
  }
}

// ---------------- batchnorm statistics ----------------

#define BN_ROWS 256
__global__ void k_bnstats(const float* __restrict__ xcat,
                          float* __restrict__ bnsum, float* __restrict__ bnsumsq) {
  int c = threadIdx.x;                        // 256 channels
  int rbase = blockIdx.x * BN_ROWS;
  float s = 0.0f, sq = 0.0f;
  for (int r = 0; r < BN_ROWS; ++r) {
    int nrow = rbase + r;
    if (nrow < NN) {
      float v = xcat[(size_t)nrow * INNER + c];
      s += v; sq += v * v;
    }
  }
  atomicAdd(&bnsum[c], s);
  atomicAdd(&bnsumsq[c], sq);
}

// ---------------- fold BN into final projection: Wp = s⊙out_w, bp = t@out_w+out_b

__global__ void k_prep(const float* __restrict__ bnsum, const float* __restrict__ bnsumsq,
                       const float* __restrict__ gamma, const float* __restrict__ beta,
                       const float* __restrict__ out_w, const float* __restrict__ out_b,
                       float* __restrict__ Wp, float* __restrict__ bp) {
  __shared__ float s_t[INNER];
  int t = threadIdx.x;                        // 256 threads
  float mu  = bnsum[t] * (1.0f / NN);
  float var = bnsumsq[t] * (1.0f / NN) - mu * mu;   // biased variance
  float sc  = gamma[t] * rsqrtf(var + BN_EPS);
  s_t[t] = beta[t] - mu * sc;
  for (int j = 0; j < D; ++j) Wp[(size_t)t * D + j] = sc * out_w[(size_t)t * D + j];
  __syncthreads();
  if (t < D) {
    float acc = out_b[t];
    for (int c = 0; c < INNER; ++c) acc += s_t[c] * out_w[(size_t)c * D + t];
    bp[t] = acc;
  }
}

// ---------------- final GEMM: out = xcat @ Wp + bp ----------------

__global__ __launch_bounds__(128) void k_final_gemm(
    const float* __restrict__ xcat, const float* __restrict__ Wp,
    const float* __restrict__ bp, float* __restrict__ out) {
  int wave = (int)((blockIdx.x * blockDim.x + threadIdx.x) >> 5);
  int lane = threadIdx.x & 31;
  if (wave >= NN / 16) return;
  int mbase = wave * 16;
  int n = lane & 15;
  int half = lane >> 4;

  v8f acc[4];
#pragma unroll
  for (int nt = 0; nt < 4; ++nt) acc[nt] = splat8(bp[nt * 16 + n]);

  const float* Abase = xcat + (size_t)(mbase + n) * INNER + 2 * half;
#pragma unroll 8
  for (int k = 0; k < 64; ++k) {             // K = 256 in steps of 4
    v2f a; a.x = Abase[4 * k]; a.y = Abase[4 * k + 1];
    int krow = 4 * k + 2 * half;
#pragma unroll
    for (int nt = 0; nt < 4; ++nt) {
      v2f b;
      b.x = Wp[(size_t)krow * D + nt * 16 + n];
      b.y = Wp[(size_t)(krow + 1) * D + nt * 16 + n];
      acc[nt] = wmma4(a, b, acc[nt]);
    }
  }

#pragma unroll
  for (int nt = 0; nt < 4; ++nt) {
    int ch = nt * 16 + n;
#pragma unroll
    for (int r = 0; r < 8; ++r)
      out[(size_t)(mbase + r + 8 * half) * D + ch] = acc[nt][r];
  }
}

// ---------------- launcher ----------------

extern "C" void kernel_launch(void* const* d_in, const int* in_sizes, int n_in,
                              void* d_out, int out_size, void* d_ws, size_t ws_size,
                              hipStream_t stream) {
  (void)in_sizes; (void)n_in; (void)out_size; (void)ws_size;
  const float* x         = (const float*)d_in[0];
  const int*   eidx      = (const int*)d_in[1];
  const float* edge_attr = (const float*)d_in[2];
  const float* lin_w     = (const float*)d_in[3];
  const float* lin_b     = (const float*)d_in[4];
  const float* root_emb  = (const float*)d_in[5];
  const float* edge_w    = (const float*)d_in[6];
  const float* edge_b    = (const float*)d_in[7];
  const float* gamma     = (const float*)d_in[8];
  const float* beta      = (const float*)d_in[9];
  const float* out_w     = (const float*)d_in[10];
  const float* out_b     = (const float*)d_in[11];
  float* out = (float*)d_out;
  const int* row = eidx;
  const int* col = eidx + NE;

  float* ws = (float*)d_ws;
  size_t off = 0;
  float* xcat    = ws + off; off += (size_t)NN * INNER;  // 25.6M f32
  float* h       = ws + off; off += (size_t)NN * D;
  float* outbuf  = ws + off; off += (size_t)NN * D;
  float* deg     = ws + off; off += NN;
  float* norm    = ws + off; off += NE;
  float* bnsum   = ws + off; off += INNER;
  float* bnsumsq = ws + off; off += INNER;
  float* Wp      = ws + off; off += (size_t)INNER * D;
  float* bp      = ws + off; off += D;

  k_init<<<(NN + 255) / 256, 256, 0, stream>>>(deg, bnsum, bnsumsq);
  k_degree<<<(NE + 255) / 256, 256, 0, stream>>>(row, deg);
  k_norm<<<(NE + 255) / 256, 256, 0, stream>>>(row, col, deg, norm);
  k_copy_x<<<(NN * D + 255) / 256, 256, 0, stream>>>(x, xcat);

  const int node_waves = NN / 16;   // 6250
  const int edge_waves = NE / 16;   // 78125
  for (int l = 0; l < NL; ++l) {
    k_node_gemm<<<(node_waves * 32 + 127) / 128, 128, 0, stream>>>(
        xcat, lin_w, lin_b, root_emb, deg, h, outbuf, l);
    k_edge<<<(edge_waves * 32 + 127) / 128, 128, 0, stream>>>(
        edge_attr, edge_w, edge_b, row, col, norm, h, outbuf, l);
    k_relu_copy<<<(NN * D + 255) / 256, 256, 0, stream>>>(outbuf, xcat, l);
  }

  k_bnstats<<<(NN + BN_ROWS - 1) / BN_ROWS, INNER, 0, stream>>>(xcat, bnsum, bnsumsq);
  k_prep<<<1, INNER, 0, stream>>>(bnsum, bnsumsq, gamma, beta, out_w, out_b, Wp, bp);
  k_final_gemm<<<(node_waves * 32 + 127) / 128, 128, 0, stream>>>(xcat, Wp, bp, out);
}